// Build_gwc_volume_unfold_63136019251721
// MI455X (gfx1250) — compile-verified
//
#include <hip/hip_runtime.h>

// GWC cost volume:  vol[b,g,d,h,w] = sum_c ref[b,g,c,h,w]*tgt[b,g,c,h,w-d]
// B=2, G=8, Cg=64, H=128, W=256, D=48.  fp32 via V_WMMA_F32_16X16X4_F32.

typedef __attribute__((ext_vector_type(2))) float v2f;
typedef __attribute__((ext_vector_type(8))) float v8f;

#define W_DIM 256
#define H_DIM 128
#define CG    64
#define MAXD  48
#define BSTRIDE 257           // 256+1: diagonal LDS scatter is bank-conflict-free

__global__ __launch_bounds__(256)
void gwc_volume_wmma_kernel(const float* __restrict__ ref,
                            const float* __restrict__ tgt,
                            float* __restrict__ out)
{
    __shared__ float band[MAXD * BSTRIDE];     // 49,344 bytes

    const int h    = blockIdx.x;               // 0..127
    const int bg   = blockIdx.y;               // b*G+g, 0..15
    const int tid  = threadIdx.x;              // 0..255 (8 waves)
    const int wave = tid >> 5;
    const int lane = tid & 31;
    const int hi   = lane >> 4;                // 0/1 : K-half select
    const int lo   = lane & 15;                // M/N index

    const size_t HW = (size_t)H_DIM * W_DIM;   // 32768
    const float* refs = ref + (size_t)bg * CG * HW + (size_t)h * W_DIM;
    const float* tgts = tgt + (size_t)bg * CG * HW + (size_t)h * W_DIM;

    // ---- compute band via WMMA: C[m][n] = sum_c ref[w0+m][c]*tgt[w1+n][c] ----
    for (int rt = wave; rt < 16; rt += 8) {    // 2 row tiles per wave
        const int w0 = rt * 16;

        // A fragments for the whole K=64 depth (16 steps of K=4), hoisted.
        // 32-bit A 16x4 layout: lane -> M=lo; VGPR0=K(2*hi), VGPR1=K(2*hi+1)
        v2f afrag[16];
#pragma unroll
        for (int kk = 0; kk < 16; ++kk) {
            const float* p = refs + (size_t)(4 * kk + 2 * hi) * HW + (w0 + lo);
            afrag[kk].x = p[0];
            afrag[kk].y = p[HW];
        }

#pragma unroll
        for (int ct = 0; ct < 4; ++ct) {
            const int w1 = w0 - 48 + 16 * ct;  // column tile base (mult of 16)
            v8f acc = {0.f, 0.f, 0.f, 0.f, 0.f, 0.f, 0.f, 0.f};

            if (w1 >= 0) {                     // w1<0 => whole tile in zero-pad
#pragma unroll
                for (int kk = 0; kk < 16; ++kk) {
                    const float* q = tgts + (size_t)(4 * kk + 2 * hi) * HW + (w1 + lo);
                    v2f bfrag;
                    bfrag.x = q[0];
                    bfrag.y = q[HW];
                    acc = __builtin_amdgcn_wmma_f32_16x16x4_f32(
                        /*neg_a=*/false, afrag[kk],
                        /*neg_b=*/false, bfrag,
                        /*c_mod=*/(short)0, acc,
                        /*reuse_a=*/false, /*reuse_b=*/false);
                }
            }

            // Scatter to LDS band.  D layout: vgpr v, lane -> M=v+8*hi, N=lo.
            // d = (w0-w1) + m - n ; w = w0 + m.  Each (d,w) written exactly once.
            const int dn = (w0 - w1) - lo;
#pragma unroll
            for (int v = 0; v < 8; ++v) {
                const int m = v + 8 * hi;
                const int d = dn + m;
                if (d >= 0 && d < MAXD) {
                    band[d * BSTRIDE + (w0 + m)] = acc[v];
                }
            }
        }
    }

    __syncthreads();

    // ---- coalesced write-out: out[((bg*48 + d)*H + h)*W + w] ----
    float* outs = out + (size_t)bg * MAXD * HW + (size_t)h * W_DIM;
    for (int i = tid; i < MAXD * W_DIM; i += 256) {
        const int d = i >> 8;                  // /256
        const int w = i & 255;
        outs[(size_t)d * HW + w] = band[d * BSTRIDE + w];
    }
}

extern "C" void kernel_launch(void* const* d_in, const int* in_sizes, int n_in,
                              void* d_out, int out_size, void* d_ws, size_t ws_size,
                              hipStream_t stream) {
    (void)in_sizes; (void)n_in; (void)d_ws; (void)ws_size; (void)out_size;
    const float* ref = (const float*)d_in[0];
    const float* tgt = (const float*)d_in[1];
    float* out = (float*)d_out;
    dim3 grid(H_DIM, 16);                      // (h, b*G+g) = 2048 workgroups
    dim3 block(256);
    gwc_volume_wmma_kernel<<<grid, block, 0, stream>>>(ref, tgt, out);
}